// DFFAdapterLayer_64733747085589
// MI455X (gfx1250) — compile-verified
//
#include <hip/hip_runtime.h>
#include <math.h>

typedef __attribute__((ext_vector_type(2))) float v2f;
typedef __attribute__((ext_vector_type(4))) float v4f;
typedef __attribute__((ext_vector_type(8))) float v8f;

#define DIM   768
#define NH    4
#define DH    192
#define NEXP  6
#define TOPK  3
#define RPAD  32
#define RTOT  24
#define BATCH 65536
#define WAVES 4
#define ZSTR  196   // z-slice LDS row stride: conflict-free, 16B-aligned rows
#define TSTR  34    // tmp-tile LDS row stride: conflict-free, 8B-aligned rows
#define AWSTR 196   // Aw LDS row stride (rows = rank cols j): conflict-free, 16B-aligned
#define BWSTR 36    // Bw LDS row stride (rows = d cols): conflict-free, 16B-aligned

// ---------------------------------------------------------------------------
// Setup: fold batch-independent routing (softmax -> top3 -> renorm) into
// per-head combined low-rank factors, stored transposed for WMMA B-fragments:
//   AwT[h][j][d] = A[e(h,j)][d][r(j)]                     (j in [0,24), pad 0)
//   BwT[h][d][j] = w(h,j) * B[e(h,j)][r(j)][d]            (j in [0,24), pad 0)
// ---------------------------------------------------------------------------
__global__ void dff_setup_kernel(const float* __restrict__ A,
                                 const float* __restrict__ Bm,
                                 const float* __restrict__ Zs,
                                 const float* __restrict__ Za,
                                 float* __restrict__ AwT,
                                 float* __restrict__ BwT)
{
    __shared__ int   s_e[2][NH][TOPK];
    __shared__ float s_w[2][NH][TOPK];

    if (threadIdx.x == 0) {
        for (int r = 0; r < 2; ++r) {
            const float* Zr = r ? Za : Zs;
            for (int h = 0; h < NH; ++h) {
                float p[NEXP];
                float m = -1e30f;
                for (int e = 0; e < NEXP; ++e) m = fmaxf(m, Zr[h * NEXP + e]);
                float s = 0.f;
                for (int e = 0; e < NEXP; ++e) { p[e] = expf(Zr[h * NEXP + e] - m); s += p[e]; }
                for (int e = 0; e < NEXP; ++e) p[e] /= s;
                bool used[NEXP] = {false, false, false, false, false, false};
                float tsum = 0.f;
                for (int k = 0; k < TOPK; ++k) {
                    int best = 0; float bv = -1.f;
                    for (int e = 0; e < NEXP; ++e)
                        if (!used[e] && p[e] > bv) { bv = p[e]; best = e; }
                    used[best] = true;
                    s_e[r][h][k] = best;
                    s_w[r][h][k] = bv;
                    tsum += bv;
                }
                const float inv = 1.f / (tsum + 1e-8f);
                for (int k = 0; k < TOPK; ++k) s_w[r][h][k] *= inv;
            }
        }
    }
    __syncthreads();

    const int tot = NH * RPAD * DH;  // 24576
    for (int i = threadIdx.x; i < tot; i += blockDim.x) {
        const int h = i / (RPAD * DH);
        const int rem = i % (RPAD * DH);
        const int j = rem / DH;
        const int d = rem % DH;
        float v = 0.f;
        if (j < RTOT) {
            const int r  = j / 12;
            const int k  = (j % 12) / 4;
            const int rr = j % 4;
            const int e  = s_e[r][h][k];
            v = A[(e * DH + d) * 4 + rr];        // A: [6,192,4]
        }
        AwT[i] = v;
    }
    for (int i = threadIdx.x; i < tot; i += blockDim.x) {
        const int h = i / (DH * RPAD);
        const int rem = i % (DH * RPAD);
        const int d = rem / RPAD;
        const int j = rem % RPAD;
        float v = 0.f;
        if (j < RTOT) {
            const int r  = j / 12;
            const int k  = (j % 12) / 4;
            const int rr = j % 4;
            const int e  = s_e[r][h][k];
            v = s_w[r][h][k] * Bm[(e * 4 + rr) * DH + d];   // B: [6,4,192]
        }
        BwT[i] = v;
    }
}

// ---------------------------------------------------------------------------
// Main: one wave per 16-token tile; per-head weight slices staged in LDS once
// per block so both WMMA inner loops are pure LDS + v_wmma. z is the only HBM
// stream (non-temporal in, non-temporal out).
// ---------------------------------------------------------------------------
__global__ __launch_bounds__(WAVES * 32) void dff_main_kernel(
    const float* __restrict__ z,
    const float* __restrict__ AwT,
    const float* __restrict__ BwT,
    float* __restrict__ out)
{
    __shared__ float zs[WAVES][16][ZSTR];  // per-wave z slice
    __shared__ float ts[WAVES][16][TSTR];  // per-wave stage-1 result
    __shared__ float aw[RPAD][AWSTR];      // block-shared Aw[h] slice: [j][d]
    __shared__ float bw[DH][BWSTR];        // block-shared Bw[h] slice: [d][j]

    const int tid   = threadIdx.x;
    const int lane  = tid & 31;
    const int wv    = tid >> 5;
    const int lrow  = lane & 15;            // M row (A frags) or N col (B/D frags)
    const int khalf = (lane >> 4) << 1;     // K sub-offset: lanes16-31 -> K+2
    const int hi8   = (lane >> 4) << 3;     // D-frag: lanes16-31 -> M+8
    const size_t tokBase = ((size_t)blockIdx.x * WAVES + wv) * 16u;

    for (int h = 0; h < NH; ++h) {
        // --- per-wave coalesced stage of 16x192 z slice (non-temporal) ---
        const float* zsrc = z + tokBase * DIM + (size_t)h * DH;
        #pragma unroll
        for (int i = lane; i < 16 * (DH / 4); i += 32) {
            const int row = i / (DH / 4);
            const int c4  = (i % (DH / 4)) * 4;
            *(v4f*)&zs[wv][row][c4] =
                __builtin_nontemporal_load((const v4f*)(zsrc + (size_t)row * DIM + c4));
        }
        // --- block-wide stage of per-head weight slices into LDS ---
        {
            const float* awg = AwT + h * RPAD * DH;
            #pragma unroll
            for (int i = tid; i < RPAD * (DH / 4); i += WAVES * 32) {
                const int row = i / (DH / 4);
                const int c4  = (i % (DH / 4)) * 4;
                *(v4f*)&aw[row][c4] = *(const v4f*)(awg + row * DH + c4);
            }
            const float* bwg = BwT + h * DH * RPAD;
            #pragma unroll
            for (int i = tid; i < DH * (RPAD / 4); i += WAVES * 32) {
                const int row = i / (RPAD / 4);
                const int c4  = (i % (RPAD / 4)) * 4;
                *(v4f*)&bw[row][c4] = *(const v4f*)(bwg + row * RPAD + c4);
            }
        }
        __syncthreads();

        // --- stage 1: tmp = z_h @ Aw_h   (M=16, K=192, N=32), all-LDS ---
        v8f acc0 = {};
        v8f acc1 = {};
        #pragma unroll 4
        for (int kk = 0; kk < DH / 4; ++kk) {
            const int k0 = kk * 4 + khalf;
            const v2f a  = *(const v2f*)&zs[wv][lrow][k0];
            const v2f b0 = *(const v2f*)&aw[lrow][k0];
            const v2f b1 = *(const v2f*)&aw[lrow + 16][k0];
            acc0 = __builtin_amdgcn_wmma_f32_16x16x4_f32(false, a, false, b0,
                                                         (short)0, acc0, false, false);
            acc1 = __builtin_amdgcn_wmma_f32_16x16x4_f32(false, a, false, b1,
                                                         (short)0, acc1, false, false);
        }

        // --- reshape D fragments -> A fragments through LDS ---
        #pragma unroll
        for (int r = 0; r < 8; ++r) {
            ts[wv][r + hi8][lrow]      = acc0[r];
            ts[wv][r + hi8][lrow + 16] = acc1[r];
        }
        __syncthreads();

        // --- stage 2: delta = tmp @ Bw_h  (M=16, K=32, N=192), fused epilogue ---
        float* outp = out + tokBase * DIM + (size_t)h * DH;
        #pragma unroll 2
        for (int t = 0; t < DH / 16; ++t) {
            const int n0 = t * 16;
            v8f acc = {};
            #pragma unroll
            for (int kk = 0; kk < RPAD / 4; ++kk) {
                const int k0 = kk * 4 + khalf;
                const v2f a = *(const v2f*)&ts[wv][lrow][k0];
                const v2f b = *(const v2f*)&bw[n0 + lrow][k0];
                acc = __builtin_amdgcn_wmma_f32_16x16x4_f32(false, a, false, b,
                                                            (short)0, acc, false, false);
            }
            #pragma unroll
            for (int r = 0; r < 8; ++r) {
                const int tok = r + hi8;
                const int col = n0 + lrow;
                __builtin_nontemporal_store(zs[wv][tok][col] + 0.5f * acc[r],
                                            &outp[(size_t)tok * DIM + col]);
            }
        }
        __syncthreads();
    }
}

extern "C" void kernel_launch(void* const* d_in, const int* in_sizes, int n_in,
                              void* d_out, int out_size, void* d_ws, size_t ws_size,
                              hipStream_t stream)
{
    (void)in_sizes; (void)n_in; (void)out_size; (void)ws_size;
    const float* z  = (const float*)d_in[0];
    const float* A  = (const float*)d_in[1];
    const float* Bm = (const float*)d_in[2];
    const float* Zs = (const float*)d_in[3];
    const float* Za = (const float*)d_in[4];
    float* out = (float*)d_out;

    float* AwT = (float*)d_ws;                 // [4][32][192] f32
    float* BwT = AwT + NH * RPAD * DH;         // [4][192][32] f32  (197 KB total)

    dff_setup_kernel<<<1, 256, 0, stream>>>(A, Bm, Zs, Za, AwT, BwT);

    const int nTiles  = BATCH / 16;            // 4096
    const int nBlocks = nTiles / WAVES;        // 1024
    dff_main_kernel<<<nBlocks, WAVES * 32, 0, stream>>>(z, AwT, BwT, out);
}